// FullTensorProduct_80736795230262
// MI455X (gfx1250) — compile-verified
//
#include <hip/hip_runtime.h>
#include <math.h>

typedef float v2f __attribute__((ext_vector_type(2)));
typedef float v8f __attribute__((ext_vector_type(8)));

// ---------------- constant path tables (sorted e3nn output order) ----------------
#define NPATH 28
__constant__ int PL1[NPATH] = {0,1,2, 0,1,1,2,2,1,2, 0,1,1,2,2,1,2,2, 0,1,2,2,1,2, 1,2,2, 2};
__constant__ int PL2[NPATH] = {0,1,2, 1,0,2,1,3,1,2, 2,1,3,0,2,2,1,3, 3,2,1,3,3,2, 3,2,3, 3};
__constant__ int PL3[NPATH] = {0,0,0, 1,1,1,1,1,1,1, 2,2,2,2,2,2,2,2, 3,3,3,3,3,3, 4,4,4, 5};
__constant__ int PCS[NPATH] = {0,1,2, 3,6,9,12,15,18,21, 24,29,34,39,44,49,54,59,
                               64,71,78,85,92,99, 106,115,124, 133};
__constant__ int XOFF[3] = {0,1,4};   // x per-u feature offsets for l=0,1,2 (9 total, pad to 12)
__constant__ int YOFF[4] = {0,1,4,9}; // y feature offsets for l=0,1,2,3 (16 total)

#define WB_FLOATS (8*1728*2)  // 27648: W in GEMM1 B-fragment layout [jpair][col=kx*144+c][e]
#define NCOL 144
#define NROWS 1728            // 12 * 144

// ---------------- complex double helpers ----------------
struct cplx { double re, im; };
__device__ inline cplx cmul(cplx a, cplx b){ return {a.re*b.re - a.im*b.im, a.re*b.im + a.im*b.re}; }
__device__ inline cplx cadd(cplx a, cplx b){ return {a.re+b.re, a.im+b.im}; }
__device__ inline cplx cconj(cplx a){ return {a.re, -a.im}; }
__device__ inline cplx cscale(cplx a, double s){ return {a.re*s, a.im*s}; }

__device__ double dfact(int n){ double r = 1.0; for(int i = 2; i <= n; i++) r *= (double)i; return r; }

// Racah CG coefficients, index ((j1+m1)*d2 + (j2+m2))*d3 + (j3+m3)
__device__ void su2_cg(int j1, int j2, int j3, double* C){
  int d2 = 2*j2+1, d3 = 2*j3+1;
  int tot = (2*j1+1)*d2*d3;
  for(int t = 0; t < tot; t++) C[t] = 0.0;
  for(int m1 = -j1; m1 <= j1; m1++) for(int m2 = -j2; m2 <= j2; m2++){
    int m3 = m1 + m2;
    if(m3 < -j3 || m3 > j3) continue;
    double pref = sqrt((2.0*j3+1.0)*dfact(j3+j1-j2)*dfact(j3-j1+j2)*dfact(j1+j2-j3)/dfact(j1+j2+j3+1));
    pref *= sqrt(dfact(j3+m3)*dfact(j3-m3)*dfact(j1-m1)*dfact(j1+m1)*dfact(j2-m2)*dfact(j2+m2));
    double s = 0.0;
    for(int k = 0; k <= j1+j2-j3; k++){
      int dd[6] = {k, j1+j2-j3-k, j1-m1-k, j2+m2-k, j3-j2+m1+k, j3-j1-m2+k};
      bool ok = true;
      for(int q = 0; q < 6; q++) if(dd[q] < 0) ok = false;
      if(!ok) continue;
      double prod = 1.0;
      for(int q = 0; q < 6; q++) prod *= dfact(dd[q]);
      s += ((k & 1) ? -1.0 : 1.0) / prod;
    }
    C[((j1+m1)*d2 + (j2+m2))*d3 + (j3+m3)] = pref * s;
  }
}

// real->complex SH basis change, q[(2l+1) x (2l+1)]
__device__ void q_r2c(int l, cplx* q){
  int d = 2*l+1;
  for(int t = 0; t < d*d; t++) q[t] = {0.0, 0.0};
  double is2 = 1.0 / sqrt(2.0);
  for(int m = -l; m < 0; m++){
    q[(l+m)*d + (l-m)] = {is2, 0.0};
    q[(l+m)*d + (l+m)] = {0.0, -is2};
  }
  q[l*d + l] = {1.0, 0.0};
  for(int m = 1; m <= l; m++){
    double sg = (m & 1) ? -1.0 : 1.0;
    q[(l+m)*d + (l+m)] = {sg*is2, 0.0};
    q[(l+m)*d + (l-m)] = {0.0, sg*is2};
  }
  cplx ph;
  switch(l & 3){ case 0: ph = {1,0}; break; case 1: ph = {0,-1}; break;
                 case 2: ph = {-1,0}; break; default: ph = {0,1}; }
  for(int t = 0; t < d*d; t++) q[t] = cmul(q[t], ph);
}

// real-basis Wigner 3j, Frobenius-normalized; Cr index (i*d2 + j)*d3 + k
__device__ void w3j_real(int l1, int l2, int l3, double* Cr){
  const int d1 = 2*l1+1, d2 = 2*l2+1, d3 = 2*l3+1;
  double Ccg[5*7*11];
  su2_cg(l1, l2, l3, Ccg);
  cplx Q1[5*5], Q2[7*7], Q3[11*11];
  q_r2c(l1, Q1); q_r2c(l2, Q2); q_r2c(l3, Q3);
  cplx T1[5*7*11], T2[5*7*11];
  for(int j = 0; j < d1; j++) for(int k = 0; k < d2; k++) for(int m = 0; m < d3; m++){
    cplx s = {0,0};
    for(int i = 0; i < d1; i++) s = cadd(s, cscale(Q1[i*d1+j], Ccg[(i*d2+k)*d3+m]));
    T1[(j*d2+k)*d3+m] = s;
  }
  for(int j = 0; j < d1; j++) for(int ll = 0; ll < d2; ll++) for(int m = 0; m < d3; m++){
    cplx s = {0,0};
    for(int k = 0; k < d2; k++) s = cadd(s, cmul(Q2[k*d2+ll], T1[(j*d2+k)*d3+m]));
    T2[(j*d2+ll)*d3+m] = s;
  }
  double nrm = 0.0;
  for(int j = 0; j < d1; j++) for(int ll = 0; ll < d2; ll++) for(int nn = 0; nn < d3; nn++){
    cplx s = {0,0};
    for(int m = 0; m < d3; m++) s = cadd(s, cmul(cconj(Q3[nn*d3+m]), T2[(j*d2+ll)*d3+m]));
    Cr[(j*d2+ll)*d3+nn] = s.re;
    nrm += s.re * s.re;
  }
  double inv = 1.0 / sqrt(nrm);
  for(int t = 0; t < d1*d2*d3; t++) Cr[t] *= inv;
}

// ---------------- init kernel: build W (B-fragment layout) + output addr tables ----------------
__global__ void tp_init(float* __restrict__ ws){
  float* Wb = ws;                        // 27648 floats
  int*   At = (int*)(ws + WB_FLOATS);    // 144 ints: 63*colstart per output column
  int*   Dt = (int*)(ws + WB_FLOATS + NCOL); // 144 ints: chunk dim (2l3+1) per column
  for(int t = threadIdx.x; t < WB_FLOATS; t += blockDim.x) Wb[t] = 0.f;
  __syncthreads();
  int q = threadIdx.x;
  if(q < NPATH){
    int l1 = PL1[q], l2 = PL2[q], l3 = PL3[q], cs = PCS[q];
    int d1 = 2*l1+1, d2 = 2*l2+1, d3 = 2*l3+1;
    double Cr[5*7*11];
    w3j_real(l1, l2, l3, Cr);
    double alpha = sqrt(2.0*l3 + 1.0);
    for(int kk = 0; kk < d3; kk++){ At[cs+kk] = 63*cs; Dt[cs+kk] = d3; }
    for(int i = 0; i < d1; i++){
      int kx = XOFF[l1] + i;                 // x-feature row 0..8
      for(int j = 0; j < d2; j++){
        int jg = YOFF[l2] + j;               // y-feature 0..15
        int p1 = jg >> 1, e = jg & 1;
        for(int kk = 0; kk < d3; kk++){
          int col = kx*NCOL + (cs + kk);     // GEMM1 flattened N index 0..1727
          Wb[(p1*NROWS + col)*2 + e] = (float)(alpha * Cr[(i*d2 + j)*d3 + kk]);
        }
      }
    }
  }
}

// ---------------- main kernel: 16 batch rows per workgroup, 4 waves ----------------
__global__ __launch_bounds__(128) void tp_main(const float* __restrict__ x,
                                               const float* __restrict__ y,
                                               const float* __restrict__ ws,
                                               float* __restrict__ out){
  __shared__ float wy[16 * NROWS];  // 110592 B: Wy[b_local][kpair p][c][e], kx = 2p+e

  const float* Wb = ws;
  const int* At = (const int*)(ws + WB_FLOATS);
  const int* Dt = (const int*)(ws + WB_FLOATS + NCOL);

  const int tid  = threadIdx.x;
  const int wave = tid >> 5;     // 0..3
  const int lane = tid & 31;
  const int n    = lane & 15;    // column within 16-wide tile
  const int h    = lane >> 4;    // lane half -> K-pair select
  const int b0   = blockIdx.x * 16;

  // per-lane output addressing tables for the 9 N-tiles
  int atv[9], dtv[9];
  #pragma unroll
  for(int nt = 0; nt < 9; nt++){ int col = nt*16 + n; atv[nt] = At[col]; dtv[nt] = Dt[col]; }

  // ---- GEMM1: Wy[16b x 1728] = Y[16b x 16j] * W[16j x 1728], f32 WMMA 16x16x4 ----
  v2f yf[4];
  {
    const float* yrow = y + (size_t)(b0 + n) * 16;   // A-row m = lane&15
    #pragma unroll
    for(int s = 0; s < 4; s++) yf[s] = *(const v2f*)(yrow + s*4 + h*2);  // j = 4s+2h, +1
  }
  for(int t = wave; t < 108; t += 4){
    int col = t*16 + n;
    v8f acc = {0.f,0.f,0.f,0.f,0.f,0.f,0.f,0.f};
    #pragma unroll
    for(int s = 0; s < 4; s++){
      v2f bf = *(const v2f*)(Wb + ((size_t)(2*s + h)*NROWS + col)*2);
      acc = __builtin_amdgcn_wmma_f32_16x16x4_f32(false, yf[s], false, bf,
                                                  (short)0, acc, false, false);
    }
    // scatter D tile into LDS in GEMM2 B-fragment order
    unsigned kx = (unsigned)col / 144u;
    unsigned c  = (unsigned)col - kx*144u;
    unsigned off = ((kx >> 1)*144u + c)*2u + (kx & 1u);
    #pragma unroll
    for(int r = 0; r < 8; r++){
      int bl = r + 8*h;                       // D row = batch-local index
      wy[bl*NROWS + off] = acc[r];
    }
  }
  __syncthreads();

  // ---- GEMM2 per batch row: out[64u x 144] = X1[64u x 12] * Wy[12 x 144] ----
  const int mt = wave;               // u-tile
  const int u  = mt*16 + n;          // A-row for this lane (both halves load same u)
  for(int bl = 0; bl < 16; bl++){
    const float* xrow = x + (size_t)(b0 + bl) * 576;
    // per-half K-pair loads: h=0 -> k{0,1},{4,5},{8,pad}; h=1 -> k{2,3},{6,7},{pad,pad}
    float v0 = xrow[h ? (64 + 3*u + 1)  : u];
    float v1 = xrow[h ? (64 + 3*u + 2)  : (64 + 3*u)];
    float v2 = xrow[h ? (256 + 5*u + 2) : (256 + 5*u)];
    float v3 = xrow[h ? (256 + 5*u + 3) : (256 + 5*u + 1)];
    float v4 = xrow[256 + 5*u + 4];
    if(h) v4 = 0.f;
    v2f af0 = {v0, v1}, af1 = {v2, v3}, af2 = {v4, 0.f};
    const float* wyb = wy + bl*NROWS;
    float* outrow = out + (size_t)(b0 + bl) * 9216;
    for(int nt = 0; nt < 9; nt++){
      int colb = nt*16 + n;
      v8f acc = {0.f,0.f,0.f,0.f,0.f,0.f,0.f,0.f};
      v2f bf0 = *(const v2f*)(wyb + ((0 + h)*144 + colb)*2);
      acc = __builtin_amdgcn_wmma_f32_16x16x4_f32(false, af0, false, bf0, (short)0, acc, false, false);
      v2f bf1 = *(const v2f*)(wyb + ((2 + h)*144 + colb)*2);
      acc = __builtin_amdgcn_wmma_f32_16x16x4_f32(false, af1, false, bf1, (short)0, acc, false, false);
      v2f bf2 = *(const v2f*)(wyb + ((4 + h)*144 + colb)*2);
      acc = __builtin_amdgcn_wmma_f32_16x16x4_f32(false, af2, false, bf2, (short)0, acc, false, false);
      // store with e3nn interleave: idx = 63*cs + u*d + col
      int idx0 = atv[nt] + colb;
      #pragma unroll
      for(int r = 0; r < 8; r++){
        int uu = mt*16 + r + 8*h;
        __builtin_nontemporal_store(acc[r], outrow + idx0 + uu*dtv[nt]);
      }
    }
  }
}

extern "C" void kernel_launch(void* const* d_in, const int* in_sizes, int n_in,
                              void* d_out, int out_size, void* d_ws, size_t ws_size,
                              hipStream_t stream){
  (void)n_in; (void)out_size; (void)ws_size;
  const float* x = (const float*)d_in[0];
  const float* y = (const float*)d_in[1];
  float* out = (float*)d_out;
  float* ws  = (float*)d_ws;
  int B = in_sizes[0] / 576;       // 20000
  tp_init<<<1, 128, 0, stream>>>(ws);
  tp_main<<<B / 16, 128, 0, stream>>>(x, y, ws, out);
}